// GCN_Class_Based_Feature_Anchored_29643864277070
// MI455X (gfx1250) — compile-verified
//
#include <hip/hip_runtime.h>
#include <hip/hip_bf16.h>

typedef __attribute__((ext_vector_type(2))) float v2f;
typedef __attribute__((ext_vector_type(8))) float v8f;

#define IN_CH 128
#define IN2   256
#define HID   128
#define OUTC  10

// ---------------------------------------------------------------------------
// Degree / normalization
// ---------------------------------------------------------------------------
__global__ __launch_bounds__(256) void k_deg_init(float* deg, int N) {
    int i = blockIdx.x * 256 + threadIdx.x;
    if (i < N) deg[i] = 1.0f;  // self loop
}

__global__ __launch_bounds__(256) void k_deg_count(const int* __restrict__ dst, float* deg, int E) {
    int e = blockIdx.x * 256 + threadIdx.x;
    if (e < E) unsafeAtomicAdd(&deg[dst[e]], 1.0f);
}

__global__ __launch_bounds__(256) void k_rsqrt(float* deg, int N) {
    int i = blockIdx.x * 256 + threadIdx.x;
    if (i < N) deg[i] = rsqrtf(deg[i]);
}

// ---------------------------------------------------------------------------
// GEMM1: h1[N,128] = concat(x - anchor, anchor)[N,256] @ W1[256,128]
// WMMA f32 16x16x4.  Block: 16 nodes, 8 waves (one 16-col slab each).
// ---------------------------------------------------------------------------
#define AS_STRIDE 258  // even (8B-aligned b64 frag loads), 258%64=2 -> conflict-free

__global__ __launch_bounds__(256) void k_gemm1(const float* __restrict__ x,
                                               const float* __restrict__ anc,
                                               const float* __restrict__ W1,
                                               float* __restrict__ h1, int N) {
    __shared__ float As[16 * AS_STRIDE];
    const int row0 = blockIdx.x * 16;
    const int tid = threadIdx.x;

    // Stage concat(x - anchor, anchor) strip, coalesced.
    for (int idx = tid; idx < 16 * IN2; idx += 256) {
        int r = idx >> 8, k = idx & 255;
        int node = row0 + r;
        float v = 0.0f;
        if (node < N) {
            v = (k < IN_CH) ? (x[(size_t)node * IN_CH + k] - anc[(size_t)node * IN_CH + k])
                            : anc[(size_t)node * IN_CH + (k - IN_CH)];
        }
        As[r * AS_STRIDE + k] = v;
    }
    __syncthreads();

    const int wave = tid >> 5, lane = tid & 31;
    const int grp = lane >> 4;      // 0: K+{0,1}, 1: K+{2,3} (ISA 16x4 f32 A layout)
    const int cidx = lane & 15;     // A: row M; B/D: col N
    const int n0 = wave * 16;

    v8f acc0 = {}, acc1 = {};
    for (int k0 = 0; k0 < IN2; k0 += 8) {
        {
            int kk = k0 + 2 * grp;
            v2f a, b;
            a.x = As[cidx * AS_STRIDE + kk];
            a.y = As[cidx * AS_STRIDE + kk + 1];
            b.x = W1[(size_t)kk * HID + n0 + cidx];
            b.y = W1[(size_t)(kk + 1) * HID + n0 + cidx];
            acc0 = __builtin_amdgcn_wmma_f32_16x16x4_f32(false, a, false, b, (short)0, acc0, false, false);
        }
        {
            int kk = k0 + 4 + 2 * grp;
            v2f a, b;
            a.x = As[cidx * AS_STRIDE + kk];
            a.y = As[cidx * AS_STRIDE + kk + 1];
            b.x = W1[(size_t)kk * HID + n0 + cidx];
            b.y = W1[(size_t)(kk + 1) * HID + n0 + cidx];
            acc1 = __builtin_amdgcn_wmma_f32_16x16x4_f32(false, a, false, b, (short)0, acc1, false, false);
        }
    }
    acc0 = acc0 + acc1;

    // D layout: VGPR v -> M=v (lanes 0-15), M=v+8 (lanes 16-31); N=cidx.
    float* outp = h1 + (size_t)(row0 + 8 * grp) * HID + n0 + cidx;
    if (row0 + 16 <= N) {
        // Full tile (always true for N % 16 == 0): unguarded store clause.
        #pragma unroll
        for (int v = 0; v < 8; ++v) outp[(size_t)v * HID] = acc0[v];
    } else {
        #pragma unroll
        for (int v = 0; v < 8; ++v) {
            if (row0 + v + 8 * grp < N) outp[(size_t)v * HID] = acc0[v];
        }
    }
}

// ---------------------------------------------------------------------------
// Self-loop init: agg1 = h1 * dinv^2   (float4 per thread)
// ---------------------------------------------------------------------------
__global__ __launch_bounds__(256) void k_self1(const float* __restrict__ h1,
                                               const float* __restrict__ dinv,
                                               float* __restrict__ agg1, int N) {
    int i = blockIdx.x * 256 + threadIdx.x;            // over N * 32 float4s
    if (i >= N * (HID / 4)) return;
    int node = i >> 5;
    float s = dinv[node]; s = s * s;
    float4 v = *(const float4*)(h1 + (size_t)i * 4);
    v.x *= s; v.y *= s; v.z *= s; v.w *= s;
    *(float4*)(agg1 + (size_t)i * 4) = v;
}

// ---------------------------------------------------------------------------
// Edge scatter 1: one wave per edge, float4 per lane, 4 fp32 atomics.
// ---------------------------------------------------------------------------
__global__ __launch_bounds__(256) void k_scatter1(const int* __restrict__ src,
                                                  const int* __restrict__ dst,
                                                  const float* __restrict__ h1,
                                                  const float* __restrict__ dinv,
                                                  float* __restrict__ agg1, int E) {
    int gt = blockIdx.x * 256 + threadIdx.x;
    int e = gt >> 5;
    int lane = gt & 31;
    if (e >= E) return;
    int s = src[e], d = dst[e];
    float nrm = dinv[s] * dinv[d];
    float4 v = *(const float4*)(h1 + (size_t)s * HID + lane * 4);
    float* o = agg1 + (size_t)d * HID + lane * 4;
    unsafeAtomicAdd(o + 0, v.x * nrm);
    unsafeAtomicAdd(o + 1, v.y * nrm);
    unsafeAtomicAdd(o + 2, v.z * nrm);
    unsafeAtomicAdd(o + 3, v.w * nrm);
}

// ---------------------------------------------------------------------------
// GEMM2: h2[N,10] = relu(agg1 + b1)[N,128] @ W2[128,10]  (padded to 16 cols)
// Block: 64 nodes, 4 waves (16 nodes each).
// ---------------------------------------------------------------------------
#define AS2_STRIDE 132  // even, 132%64=4 -> conflict-free across 16 rows

__global__ __launch_bounds__(128) void k_gemm2(const float* __restrict__ agg1,
                                               const float* __restrict__ b1,
                                               const float* __restrict__ W2,
                                               float* __restrict__ h2, int N) {
    __shared__ float As[64 * AS2_STRIDE];
    __shared__ float Bs[HID * 16];
    const int node0 = blockIdx.x * 64;
    const int tid = threadIdx.x;

    for (int idx = tid; idx < 64 * HID; idx += 128) {
        int r = idx >> 7, k = idx & 127;
        int node = node0 + r;
        float v = 0.0f;
        if (node < N) {
            v = agg1[(size_t)node * HID + k] + b1[k];
            v = v > 0.0f ? v : 0.0f;  // relu
        }
        As[r * AS2_STRIDE + k] = v;
    }
    for (int idx = tid; idx < HID * 16; idx += 128) {
        int k = idx >> 4, n = idx & 15;
        Bs[idx] = (n < OUTC) ? W2[(size_t)k * OUTC + n] : 0.0f;
    }
    __syncthreads();

    const int wave = tid >> 5, lane = tid & 31;
    const int grp = lane >> 4, cidx = lane & 15;
    const int rbase = wave * 16;

    v8f acc0 = {}, acc1 = {};
    for (int k0 = 0; k0 < HID; k0 += 8) {
        {
            int kk = k0 + 2 * grp;
            v2f a, b;
            a.x = As[(rbase + cidx) * AS2_STRIDE + kk];
            a.y = As[(rbase + cidx) * AS2_STRIDE + kk + 1];
            b.x = Bs[kk * 16 + cidx];
            b.y = Bs[(kk + 1) * 16 + cidx];
            acc0 = __builtin_amdgcn_wmma_f32_16x16x4_f32(false, a, false, b, (short)0, acc0, false, false);
        }
        {
            int kk = k0 + 4 + 2 * grp;
            v2f a, b;
            a.x = As[(rbase + cidx) * AS2_STRIDE + kk];
            a.y = As[(rbase + cidx) * AS2_STRIDE + kk + 1];
            b.x = Bs[kk * 16 + cidx];
            b.y = Bs[(kk + 1) * 16 + cidx];
            acc1 = __builtin_amdgcn_wmma_f32_16x16x4_f32(false, a, false, b, (short)0, acc1, false, false);
        }
    }
    acc0 = acc0 + acc1;

    const int mrow0 = node0 + rbase + 8 * grp;
    float* outp = h2 + (size_t)mrow0 * OUTC + cidx;
    if (node0 + 64 <= N) {
        // Full tile: single lane-mask predicate, unguarded address math.
        if (cidx < OUTC) {
            #pragma unroll
            for (int v = 0; v < 8; ++v) outp[(size_t)v * OUTC] = acc0[v];
        }
    } else {
        #pragma unroll
        for (int v = 0; v < 8; ++v) {
            if (mrow0 + v < N && cidx < OUTC) outp[(size_t)v * OUTC] = acc0[v];
        }
    }
}

// ---------------------------------------------------------------------------
// Self-loop + bias: out = h2 * dinv^2 + b2
// ---------------------------------------------------------------------------
__global__ __launch_bounds__(256) void k_self2(const float* __restrict__ h2,
                                               const float* __restrict__ dinv,
                                               const float* __restrict__ b2,
                                               float* __restrict__ out, int N) {
    int i = blockIdx.x * 256 + threadIdx.x;
    if (i >= N * OUTC) return;
    int node = i / OUTC;
    int c = i - node * OUTC;
    float s = dinv[node];
    out[i] = h2[i] * s * s + b2[c];
}

// ---------------------------------------------------------------------------
// Edge scatter 2: 16 lanes per edge, channels 0..9 active.
// ---------------------------------------------------------------------------
__global__ __launch_bounds__(256) void k_scatter2(const int* __restrict__ src,
                                                  const int* __restrict__ dst,
                                                  const float* __restrict__ h2,
                                                  const float* __restrict__ dinv,
                                                  float* __restrict__ out, int E) {
    int gt = blockIdx.x * 256 + threadIdx.x;
    int e = gt >> 4;
    int c = gt & 15;
    if (e >= E || c >= OUTC) return;
    int s = src[e], d = dst[e];
    float nrm = dinv[s] * dinv[d];
    unsafeAtomicAdd(&out[(size_t)d * OUTC + c], h2[(size_t)s * OUTC + c] * nrm);
}

// ---------------------------------------------------------------------------
// Launch
// ---------------------------------------------------------------------------
extern "C" void kernel_launch(void* const* d_in, const int* in_sizes, int n_in,
                              void* d_out, int out_size, void* d_ws, size_t ws_size,
                              hipStream_t stream) {
    const float* x    = (const float*)d_in[0];
    const int*   ei   = (const int*)d_in[1];   // [2, E] flattened
    // d_in[2] = labels (unused)
    const float* anc  = (const float*)d_in[3];
    const float* W1   = (const float*)d_in[4];
    const float* b1   = (const float*)d_in[5];
    const float* W2   = (const float*)d_in[6];
    const float* b2   = (const float*)d_in[7];
    float* out = (float*)d_out;

    const int N = in_sizes[0] / IN_CH;
    const int E = in_sizes[1] / 2;
    const int* src = ei;
    const int* dst = ei + E;

    float* h1   = (float*)d_ws;                       // N*128
    float* agg1 = h1 + (size_t)N * HID;               // N*128
    float* h2   = agg1 + (size_t)N * HID;             // N*10
    float* dinv = h2 + (size_t)N * OUTC;              // N  (deg -> dinv in place)

    // 1-3: degrees -> dinv
    k_deg_init <<<(N + 255) / 256, 256, 0, stream>>>(dinv, N);
    k_deg_count<<<(E + 255) / 256, 256, 0, stream>>>(dst, dinv, E);
    k_rsqrt    <<<(N + 255) / 256, 256, 0, stream>>>(dinv, N);

    // 4: GEMM1 (WMMA f32)
    k_gemm1<<<(N + 15) / 16, 256, 0, stream>>>(x, anc, W1, h1, N);

    // 5-6: aggregation layer 1
    k_self1   <<<(N * (HID / 4) + 255) / 256, 256, 0, stream>>>(h1, dinv, agg1, N);
    k_scatter1<<<((size_t)E * 32 + 255) / 256, 256, 0, stream>>>(src, dst, h1, dinv, agg1, E);

    // 7: GEMM2 (WMMA f32, relu+bias fused into A staging)
    k_gemm2<<<(N + 63) / 64, 128, 0, stream>>>(agg1, b1, W2, h2, N);

    // 8-9: aggregation layer 2 -> d_out
    k_self2   <<<(N * OUTC + 255) / 256, 256, 0, stream>>>(h2, dinv, b2, out, N);
    k_scatter2<<<((size_t)E * 16 + 255) / 256, 256, 0, stream>>>(src, dst, h2, dinv, out, E);
}